// GATLayer_29764123361882
// MI455X (gfx1250) — compile-verified
//
#include <hip/hip_runtime.h>
#include <hip/hip_bf16.h>
#include <math.h>

typedef __attribute__((ext_vector_type(2))) float v2f;
typedef __attribute__((ext_vector_type(8))) float v8f;

#define NEG_SLOPE_ATT 0.2f
#define NEG_SLOPE_ACT 0.01f

__device__ __forceinline__ float lrelu(float v, float s) { return v > 0.f ? v : v * s; }

// Monotonic-int trick for float atomic max (init value must be -inf).
__device__ __forceinline__ void atomicMaxF(float* a, float v) {
    if (v >= 0.f) atomicMax((int*)a, __float_as_int(v));
    else          atomicMin((unsigned int*)a, __float_as_uint(v));
}

// ---------------------------------------------------------------------------
// Kernel 1: h[N,256] = x[N,64] @ W[256,64]^T  via v_wmma_f32_16x16x4_f32.
// Block = 512 threads = 16 waves; wave w computes the 16x16 tile at cols
// [16w, 16w+16) for the 16-node row tile of this block.
// A-frag (16x4 f32): lane&15 = M row; lanes 0-15 hold K={0,1}, 16-31 K={2,3}.
// B-frag (4x16 f32): mirrored (lane&15 = N col, same K split).  B[k][n]=W[n][k].
// C/D: vgpr v, lanes 0-15 -> M=v, N=lane; lanes 16-31 -> M=v+8, N=lane-16.
// ---------------------------------------------------------------------------
__global__ void __launch_bounds__(512) gat_gemm_wmma(
    const float* __restrict__ x, const float* __restrict__ W,
    float* __restrict__ h, int N) {
    const int lane = threadIdx.x & 31;
    const int wave = threadIdx.x >> 5;           // 0..15 -> column tile
    const int row0 = blockIdx.x << 4;
    const int col0 = wave << 4;
    const int r    = lane & 15;
    const int koff = (lane >> 4) << 1;           // 0 or 2

    int arow = row0 + r; if (arow >= N) arow = N - 1;   // clamp (EXEC stays full)
    const float* __restrict__ xa = x + (size_t)arow * 64 + koff;
    const float* __restrict__ wb = W + (size_t)(col0 + r) * 64 + koff;

    v8f c = {};
#pragma unroll
    for (int k = 0; k < 64; k += 4) {
        v2f a, b;
        a[0] = xa[k]; a[1] = xa[k + 1];
        b[0] = wb[k]; b[1] = wb[k + 1];
        c = __builtin_amdgcn_wmma_f32_16x16x4_f32(false, a, false, b,
                                                  (short)0, c, false, false);
    }

    const int n  = lane & 15;
    const int mb = (lane >> 4) << 3;             // 0 or 8
#pragma unroll
    for (int v = 0; v < 8; ++v) {
        int row = row0 + mb + v;
        if (row < N) h[(size_t)row * 256 + col0 + n] = c[v];
    }
}

// ---------------------------------------------------------------------------
// Kernel 2: a_src[n,h] = <h[n,h,:], att_src[h,:]>, same for a_dst.
// ---------------------------------------------------------------------------
__global__ void gat_att(const float* __restrict__ h,
                        const float* __restrict__ att_src,
                        const float* __restrict__ att_dst,
                        float* __restrict__ a_src, float* __restrict__ a_dst,
                        int NH) {
    int i = blockIdx.x * blockDim.x + threadIdx.x;
    if (i >= NH) return;
    int hh = i & 3;
    const float* __restrict__ hp = h + (size_t)(i >> 2) * 256 + hh * 64;
    const float* __restrict__ as = att_src + hh * 64;
    const float* __restrict__ ad = att_dst + hh * 64;
    float s = 0.f, d = 0.f;
#pragma unroll 8
    for (int c = 0; c < 64; ++c) { float v = hp[c]; s += v * as[c]; d += v * ad[c]; }
    a_src[i] = s;
    a_dst[i] = d;
}

// ---------------------------------------------------------------------------
// Kernel 3: init accumulators.
// ---------------------------------------------------------------------------
__global__ void gat_init(float* __restrict__ acc, float* __restrict__ amax,
                         float* __restrict__ denom, int N) {
    int i = blockIdx.x * blockDim.x + threadIdx.x;
    if (i < N * 64) acc[i] = 0.f;
    if (i < N * 4) { amax[i] = -INFINITY; denom[i] = 0.f; }
}

// ---------------------------------------------------------------------------
// Kernel 4: segment max of leaky_relu(a_src[src]+a_dst[dst]) over dst.
// Edges e < E come from edge_index; e >= E are self-loops.
// ---------------------------------------------------------------------------
__global__ void gat_alpha_max(const int* __restrict__ ei,
                              const float* __restrict__ a_src,
                              const float* __restrict__ a_dst,
                              float* __restrict__ amax, int E, int N) {
    int tot = E + N;
    for (int e = blockIdx.x * blockDim.x + threadIdx.x; e < tot;
         e += gridDim.x * blockDim.x) {
        int s, d;
        if (e < E) { s = ei[e]; d = ei[E + e]; } else { s = d = e - E; }
#pragma unroll
        for (int hh = 0; hh < 4; ++hh) {
            float v = lrelu(a_src[s * 4 + hh] + a_dst[d * 4 + hh], NEG_SLOPE_ATT);
            atomicMaxF(&amax[d * 4 + hh], v);
        }
    }
}

// ---------------------------------------------------------------------------
// Kernel 5: denom[dst,h] += exp(alpha - amax[dst,h]).
// ---------------------------------------------------------------------------
__global__ void gat_alpha_sum(const int* __restrict__ ei,
                              const float* __restrict__ a_src,
                              const float* __restrict__ a_dst,
                              const float* __restrict__ amax,
                              float* __restrict__ denom, int E, int N) {
    int tot = E + N;
    for (int e = blockIdx.x * blockDim.x + threadIdx.x; e < tot;
         e += gridDim.x * blockDim.x) {
        int s, d;
        if (e < E) { s = ei[e]; d = ei[E + e]; } else { s = d = e - E; }
#pragma unroll
        for (int hh = 0; hh < 4; ++hh) {
            float v = lrelu(a_src[s * 4 + hh] + a_dst[d * 4 + hh], NEG_SLOPE_ATT);
            atomicAdd(&denom[d * 4 + hh], __expf(v - amax[d * 4 + hh]));
        }
    }
}

// ---------------------------------------------------------------------------
// Kernel 6: wave-per-edge scatter.  Folds the head mean in-register:
// acc[dst,c] += (1/H) * sum_h alpha[e,h] * hfeat[src,h,c]  -> 64 atomics/edge.
// Lanes cover c = lane and c = lane+32.
// ---------------------------------------------------------------------------
__global__ void __launch_bounds__(256) gat_scatter(
    const int* __restrict__ ei, const float* __restrict__ a_src,
    const float* __restrict__ a_dst, const float* __restrict__ amax,
    const float* __restrict__ denom, const float* __restrict__ h,
    float* __restrict__ acc, int E, int N) {
    int tot  = E + N;
    int lane = threadIdx.x & 31;
    int wave = blockIdx.x * (blockDim.x >> 5) + (threadIdx.x >> 5);
    int nw   = gridDim.x * (blockDim.x >> 5);
    for (int e = wave; e < tot; e += nw) {
        int s, d;
        if (e < E) { s = ei[e]; d = ei[E + e]; } else { s = d = e - E; }
        float al[4];
#pragma unroll
        for (int hh = 0; hh < 4; ++hh) {
            float v = lrelu(a_src[s * 4 + hh] + a_dst[d * 4 + hh], NEG_SLOPE_ATT);
            al[hh] = __expf(v - amax[d * 4 + hh]) /
                     (denom[d * 4 + hh] + 1e-16f) * 0.25f;   // *1/H head mean
        }
        const float* __restrict__ hp = h + (size_t)s * 256;
#pragma unroll
        for (int part = 0; part < 2; ++part) {
            int c = lane + part * 32;
            float m = al[0] * hp[c]       + al[1] * hp[64 + c] +
                      al[2] * hp[128 + c] + al[3] * hp[192 + c];
            atomicAdd(&acc[(size_t)d * 64 + c], m);
        }
    }
}

// ---------------------------------------------------------------------------
// Kernel 7: out = leaky_relu(acc + bias, 0.01) + x
// ---------------------------------------------------------------------------
__global__ void gat_final(const float* __restrict__ acc,
                          const float* __restrict__ bias,
                          const float* __restrict__ x,
                          float* __restrict__ out, int NC) {
    int i = blockIdx.x * blockDim.x + threadIdx.x;
    if (i >= NC) return;
    float t = acc[i] + bias[i & 63];
    t = t > 0.f ? t : NEG_SLOPE_ACT * t;
    out[i] = t + x[i];
}

extern "C" void kernel_launch(void* const* d_in, const int* in_sizes, int n_in,
                              void* d_out, int out_size, void* d_ws, size_t ws_size,
                              hipStream_t stream) {
    const float* x       = (const float*)d_in[0];   // [N,64]
    const int*   ei      = (const int*)d_in[1];     // [2,E]
    const float* W       = (const float*)d_in[2];   // [256,64]
    const float* att_src = (const float*)d_in[3];   // [4,64]
    const float* att_dst = (const float*)d_in[4];   // [4,64]
    const float* bias    = (const float*)d_in[5];   // [64]

    const int N = in_sizes[0] / 64;
    const int E = in_sizes[1] / 2;

    // Workspace carve-up (floats).
    float* ws    = (float*)d_ws;
    float* h     = ws;                       // N*256
    float* acc   = h + (size_t)N * 256;      // N*64
    float* a_src = acc + (size_t)N * 64;     // N*4
    float* a_dst = a_src + (size_t)N * 4;    // N*4
    float* amax  = a_dst + (size_t)N * 4;    // N*4
    float* denom = amax + (size_t)N * 4;     // N*4

    float* out = (float*)d_out;              // [N,64]

    // 1. GEMM via WMMA: 16-row tiles, 16 waves/block (one 16x16 tile each).
    gat_gemm_wmma<<<(N + 15) / 16, 512, 0, stream>>>(x, W, h, N);

    // 2. Attention coefficients per (node, head).
    gat_att<<<(N * 4 + 255) / 256, 256, 0, stream>>>(h, att_src, att_dst,
                                                     a_src, a_dst, N * 4);

    // 3. Init accumulators.
    gat_init<<<(N * 64 + 255) / 256, 256, 0, stream>>>(acc, amax, denom, N);

    // 4-5. Segment softmax statistics over destination nodes.
    gat_alpha_max<<<2048, 256, 0, stream>>>(ei, a_src, a_dst, amax, E, N);
    gat_alpha_sum<<<2048, 256, 0, stream>>>(ei, a_src, a_dst, amax, denom, E, N);

    // 6. Wave-per-edge weighted scatter with in-register head fold.
    gat_scatter<<<2048, 256, 0, stream>>>(ei, a_src, a_dst, amax, denom,
                                          h, acc, E, N);

    // 7. Bias + activation + residual.
    gat_final<<<(N * 64 + 255) / 256, 256, 0, stream>>>(acc, bias, x, out, N * 64);
}